// GraphAttentionNetwork_57793079935294
// MI455X (gfx1250) — compile-verified
//
#include <hip/hip_runtime.h>
#include <math.h>

typedef __attribute__((ext_vector_type(2))) float v2f;
typedef __attribute__((ext_vector_type(8))) float v8f;

#define HEADS 4
#define CH 16
#define HC 64        // HEADS*CH
#define INCH 128
#define NEG_SLOPE 0.2f

// Order-preserving float -> unsigned map for atomicMax on floats.
__device__ __forceinline__ unsigned enc_f32(float f) {
    unsigned u = __float_as_uint(f);
    return (u & 0x80000000u) ? ~u : (u | 0x80000000u);
}
__device__ __forceinline__ float dec_f32(unsigned u) {
    u = (u & 0x80000000u) ? (u & 0x7fffffffu) : ~u;
    return __uint_as_float(u);
}

// ---------------------------------------------------------------------------
// h = x @ W  via V_WMMA_F32_16X16X4_F32.  One block = 16 rows x 64 cols,
// 4 waves, wave w owns columns [16w, 16w+16).  32 WMMAs per tile (K=128).
// ---------------------------------------------------------------------------
__global__ void __launch_bounds__(128) gemm_wmma_kernel(
    const float* __restrict__ x, const float* __restrict__ W,
    float* __restrict__ h, int n)
{
    __shared__ float sX[16][INCH + 1];   // pad -> conflict-free column reads
    __shared__ float sW[INCH][HC + 1];   // pad -> kills 2-way row conflict
    const int tid = threadIdx.x;
    const int m0  = blockIdx.x * 16;

    for (int i = tid; i < 16 * INCH; i += 128) {
        int r = i >> 7, k = i & 127;
        int row = m0 + r; if (row >= n) row = n - 1;   // clamp (store predicated)
        sX[r][k] = x[(size_t)row * INCH + k];
    }
    for (int i = tid; i < INCH * HC; i += 128) {
        sW[i >> 6][i & 63] = W[i];
    }
    __syncthreads();

    const int wave = tid >> 5;
    const int lane = tid & 31;
    const int mr   = lane & 15;            // row for A, col for B/C/D
    const int koff = (lane >> 4) << 1;     // lanes 16-31 hold K+2,K+3
    const int col  = (wave << 4) + mr;     // global output column

    v8f acc = {0.f, 0.f, 0.f, 0.f, 0.f, 0.f, 0.f, 0.f};
#pragma unroll
    for (int k = 0; k < INCH; k += 4) {
        v2f a, b;
        a.x = sX[mr][k + koff];
        a.y = sX[mr][k + koff + 1];
        b.x = sW[k + koff][col];
        b.y = sW[k + koff + 1][col];
        acc = __builtin_amdgcn_wmma_f32_16x16x4_f32(
            false, a, false, b, (short)0, acc, false, false);
    }
    // C/D layout: VGPR r -> M=r (lanes 0-15), M=r+8 (lanes 16-31); N = lane&15
#pragma unroll
    for (int r = 0; r < 8; ++r) {
        int row = m0 + r + ((lane >> 4) << 3);
        if (row < n) h[(size_t)row * HC + col] = acc[r];
    }
}

// a_src[n,h] = <h[n,h,:], att_src[h,:]> ; same for a_dst (single pass over h)
__global__ void att_coef_kernel(const float* __restrict__ h,
    const float* __restrict__ att_src, const float* __restrict__ att_dst,
    float* __restrict__ a_src, float* __restrict__ a_dst, int n)
{
    int idx = blockIdx.x * blockDim.x + threadIdx.x;
    if (idx >= n * HEADS) return;
    int head = idx & 3;
    int node = idx >> 2;
    const float* hv = h + (size_t)node * HC + head * CH;
    float s = 0.f, d = 0.f;
#pragma unroll
    for (int c = 0; c < CH; ++c) {
        float v = hv[c];
        s += v * att_src[head * CH + c];
        d += v * att_dst[head * CH + c];
    }
    a_src[idx] = s;
    a_dst[idx] = d;
}

// Pass 1 over edges: segment max of leaky_relu(a_src[s]+a_dst[d]) into e_max[d]
__global__ void edge_max_kernel(const int* __restrict__ srcv,
    const int* __restrict__ dstv,
    const float* __restrict__ a_src, const float* __restrict__ a_dst,
    unsigned* __restrict__ emax, int E, int n)
{
    int idx = blockIdx.x * blockDim.x + threadIdx.x;
    int total = (E + n) * HEADS;
    if (idx >= total) return;
    int head = idx & 3;
    int e_id = idx >> 2;
    int s, d;
    if (e_id < E) { s = srcv[e_id]; d = dstv[e_id]; } else { s = d = e_id - E; }
    float e = a_src[s * HEADS + head] + a_dst[d * HEADS + head];
    e = (e >= 0.f) ? e : NEG_SLOPE * e;
    atomicMax(&emax[d * HEADS + head], enc_f32(e));
}

// Pass 2: p = exp(e - max); denom[d] += p; out[d,:] += p*h[s,:]; stash p as alpha.
// 16 lanes per (edge,head): lane c owns channel c -> coalesced 64B h gather,
// one out global_atomic_add_f32 per lane; lane 0 also adds to denom.
__global__ void edge_agg_kernel(const int* __restrict__ srcv,
    const int* __restrict__ dstv,
    const float* __restrict__ a_src, const float* __restrict__ a_dst,
    const unsigned* __restrict__ emax, const float* __restrict__ h,
    float* __restrict__ denom, float* __restrict__ out_accum,
    float* __restrict__ alpha_exp, int E, int n)
{
    long long idx = (long long)blockIdx.x * blockDim.x + threadIdx.x;
    long long total = (long long)(E + n) * HEADS * CH;
    if (idx >= total) return;
    int c = (int)(idx & 15);
    long long pair = idx >> 4;
    int head = (int)(pair & 3);
    int e_id = (int)(pair >> 2);
    int s, d;
    if (e_id < E) { s = srcv[e_id]; d = dstv[e_id]; } else { s = d = e_id - E; }
    float e = a_src[s * HEADS + head] + a_dst[d * HEADS + head];
    e = (e >= 0.f) ? e : NEG_SLOPE * e;
    float p = __expf(e - dec_f32(emax[d * HEADS + head]));
    if (c == 0) {
        atomicAdd(&denom[d * HEADS + head], p);
        alpha_exp[(size_t)e_id * HEADS + head] = p;
    }
    float hv = h[(size_t)s * HC + head * CH + c];
    atomicAdd(&out_accum[(size_t)d * HC + head * CH + c], p * hv);
}

// Node finalize: out = elu(accum/(denom+eps) + bias)
__global__ void node_final_kernel(float* __restrict__ out,
    const float* __restrict__ denom, const float* __restrict__ bias, int n)
{
    int idx = blockIdx.x * blockDim.x + threadIdx.x;
    if (idx >= n * HC) return;
    int node = idx >> 6;
    int hc = idx & 63;
    int head = hc >> 4;
    float v = out[idx] / (denom[node * HEADS + head] + 1e-16f) + bias[hc];
    out[idx] = (v > 0.f) ? v : (__expf(v) - 1.f);
}

// Edge finalize: alpha = exp_stored / (denom[d]+eps)
__global__ void alpha_norm_kernel(float* __restrict__ alpha,
    const float* __restrict__ denom, const int* __restrict__ dstv, int E, int n)
{
    int idx = blockIdx.x * blockDim.x + threadIdx.x;
    int total = (E + n) * HEADS;
    if (idx >= total) return;
    int head = idx & 3;
    int e_id = idx >> 2;
    int d = (e_id < E) ? dstv[e_id] : (e_id - E);
    alpha[idx] = alpha[idx] / (denom[d * HEADS + head] + 1e-16f);
}

extern "C" void kernel_launch(void* const* d_in, const int* in_sizes, int n_in,
                              void* d_out, int out_size, void* d_ws, size_t ws_size,
                              hipStream_t stream) {
    const float* x          = (const float*)d_in[0];
    const int*   edge_index = (const int*)d_in[1];
    const float* W          = (const float*)d_in[2];
    const float* att_src    = (const float*)d_in[3];
    const float* att_dst    = (const float*)d_in[4];
    const float* bias       = (const float*)d_in[5];

    const int N = in_sizes[0] / INCH;
    const int E = in_sizes[1] / 2;
    const int* srcv = edge_index;
    const int* dstv = edge_index + E;

    float* out   = (float*)d_out;                 // [N, 64]
    float* alpha = out + (size_t)N * HC;          // [(E+N), 4]

    float*    h     = (float*)d_ws;                              // [N, 64]
    float*    a_src = h + (size_t)N * HC;                        // [N, 4]
    float*    a_dst = a_src + (size_t)N * HEADS;                 // [N, 4]
    unsigned* emax  = (unsigned*)(a_dst + (size_t)N * HEADS);    // [N, 4]
    float*    denom = (float*)(emax + (size_t)N * HEADS);        // [N, 4]

    hipMemsetAsync(out,   0, sizeof(float)    * (size_t)N * HC,    stream);
    hipMemsetAsync(emax,  0, sizeof(unsigned) * (size_t)N * HEADS, stream);
    hipMemsetAsync(denom, 0, sizeof(float)    * (size_t)N * HEADS, stream);

    gemm_wmma_kernel<<<(N + 15) / 16, 128, 0, stream>>>(x, W, h, N);
    att_coef_kernel<<<(N * HEADS + 255) / 256, 256, 0, stream>>>(
        h, att_src, att_dst, a_src, a_dst, N);

    int EH = (E + N) * HEADS;
    edge_max_kernel<<<(EH + 255) / 256, 256, 0, stream>>>(
        srcv, dstv, a_src, a_dst, emax, E, N);

    long long tot = (long long)EH * CH;
    edge_agg_kernel<<<(unsigned)((tot + 255) / 256), 256, 0, stream>>>(
        srcv, dstv, a_src, a_dst, emax, h, denom, out, alpha, E, N);

    node_final_kernel<<<(N * HC + 255) / 256, 256, 0, stream>>>(
        out, denom, bias, N);
    alpha_norm_kernel<<<(EH + 255) / 256, 256, 0, stream>>>(
        alpha, denom, dstv, E, N);
}